// _AvgResNet2_11398843203605
// MI455X (gfx1250) — compile-verified
//
#include <hip/hip_runtime.h>
#include <hip/hip_bf16.h>
#include <math.h>

// ---------------- problem constants ----------------
#define BGRAPH 64
#define NNODES 4096
#define FDIM   128
#define MTOT   (BGRAPH * NNODES)      // 262144 rows
#define CHUNKS 8                      // reduce blocks per graph
#define RBLKS  (BGRAPH * CHUNKS)      // 512 reduce blocks
#define RROWS  (NNODES / CHUNKS)      // 512 rows per reduce block
#define EPS    1e-5f

// ---------------- WMMA types ----------------
typedef float v2f __attribute__((ext_vector_type(2)));
typedef float v8f __attribute__((ext_vector_type(8)));

// GEMM LDS layout (floats)
#define W_STRIDE 288                  // 256 data + 32 pad -> half-waves in disjoint bank windows
#define W_WORDS  (64 * W_STRIDE)      // 18432 (64 k-pair rows)
#define Y_STRIDE 132                  // 128 data + 4 pad -> rows shift banks by 4
#define Y_WORDS  (128 * Y_STRIDE)     // 16896
#define SMEM_FLOATS (W_WORDS + Y_WORDS + 128)
#define SMEM_BYTES  (SMEM_FLOATS * 4) // 141824 B < 320 KB/WGP

__device__ __forceinline__ float elu(float v) {
    return v > 0.0f ? v : (expf(v) - 1.0f);
}

// ============ Pass A: elu + deterministic partial sums ============
// grid = 512 blocks (graph-major: rows blk*512 .. +511), block = 256
__global__ __launch_bounds__(256)
void reduce_kernel(const float* __restrict__ x, const float* __restrict__ mask,
                   float* __restrict__ S_part, float* __restrict__ cs_part,
                   float* __restrict__ cq_part, float* __restrict__ mk_part,
                   int compute_msum)
{
    const int blk = blockIdx.x;
    const int t   = threadIdx.x;
    const int col = t & 127;
    const int par = t >> 7;
    const int base = blk * RROWS;

    float sm = 0.f, cs = 0.f, cq = 0.f, mks = 0.f;
    for (int i = 0; i < RROWS / 2; ++i) {
        const int row = base + par + 2 * i;
        float v  = x[row * FDIM + col];
        float y  = elu(v);
        float mk = mask[row];
        sm += y * mk;
        cs += y;
        cq += y * y;
        mks += mk;                     // only t==0/t==128 values are used
    }

    __shared__ float red[256];
    red[t] = sm;  __syncthreads();
    if (t < 128) S_part[blk * 128 + t] = red[t] + red[t + 128];
    __syncthreads();
    red[t] = cs;  __syncthreads();
    if (t < 128) cs_part[blk * 128 + t] = red[t] + red[t + 128];
    __syncthreads();
    red[t] = cq;  __syncthreads();
    if (t < 128) cq_part[blk * 128 + t] = red[t] + red[t + 128];
    if (compute_msum) {
        __syncthreads();
        red[t] = mks; __syncthreads();
        if (t == 0) mk_part[blk] = red[0] + red[128];
    }
}

// ============ Stats finalize: glob_avg + BN scale/shift ============
// 1 block x 256
__global__ __launch_bounds__(256)
void prep_kernel(const float* __restrict__ S_part, const float* __restrict__ cs_part,
                 const float* __restrict__ cq_part, const float* __restrict__ mk_part,
                 const float* __restrict__ gamma, const float* __restrict__ beta,
                 float* __restrict__ glob, float* __restrict__ scale,
                 float* __restrict__ shiftv)
{
    __shared__ float msum_s[BGRAPH];
    __shared__ float glob_s[BGRAPH * FDIM];
    const int t = threadIdx.x;

    if (t < BGRAPH) {
        float s = 0.f;
        for (int j = 0; j < CHUNKS; ++j) s += mk_part[t * CHUNKS + j];
        msum_s[t] = s;
    }
    __syncthreads();

    for (int idx = t; idx < BGRAPH * FDIM; idx += 256) {
        const int b = idx >> 7, f = idx & 127;
        float s = 0.f;
        for (int j = 0; j < CHUNKS; ++j) s += S_part[(b * CHUNKS + j) * 128 + f];
        const float g = s / msum_s[b];
        glob_s[idx] = g;
        glob[idx]   = g;
    }
    __syncthreads();

    if (t < 128) {                      // low channels: stats over all rows
        const int c = t;
        float cs = 0.f, cq = 0.f;
        for (int blk = 0; blk < RBLKS; ++blk) {
            cs += cs_part[blk * 128 + c];
            cq += cq_part[blk * 128 + c];
        }
        const float inv  = 1.0f / (float)MTOT;
        const float mean = cs * inv;
        const float var  = cq * inv - mean * mean;
        const float sc   = gamma[c] * rsqrtf(var + EPS);
        scale[c]  = sc;
        shiftv[c] = beta[c] - sc * mean;
    } else {                            // high channels: stats over the 64 graph means
        const int f = t - 128;
        float s = 0.f, sq = 0.f;
        for (int b = 0; b < BGRAPH; ++b) {
            const float g = glob_s[b * 128 + f];
            s += g; sq += g * g;
        }
        const float mean = s * (1.0f / BGRAPH);
        const float var  = sq * (1.0f / BGRAPH) - mean * mean;
        const float sc   = gamma[128 + f] * rsqrtf(var + EPS);
        scale[128 + f]  = sc;
        shiftv[128 + f] = beta[128 + f] - sc * mean;
    }
}

// ============ Fold BN+bias+glob into packed weights / row bias ============
// grid = 65 blocks x 256: blocks 0..63 -> rowbias[b][:], block 64 -> Wpk
__global__ __launch_bounds__(256)
void weights_kernel(const float* __restrict__ w, const float* __restrict__ bias,
                    const float* __restrict__ scale, const float* __restrict__ shiftv,
                    const float* __restrict__ glob, float* __restrict__ Wpk,
                    float* __restrict__ rowbias)
{
    const int t = threadIdx.x;
    if (blockIdx.x == BGRAPH) {
        // Wpk[(k/2)*256 + n*2 + (k&1)] = w[n][k] * scale[k]  (k<128 low half)
        for (int idx = t; idx < 64 * 256; idx += 256) {
            const int k2 = idx >> 8, rem = idx & 255;
            const int n = rem >> 1, k = k2 * 2 + (rem & 1);
            Wpk[idx] = w[n * 256 + k] * scale[k];
        }
    } else {
        const int b = blockIdx.x;
        if (t < 128) {
            const int o = t;
            const float* wrow = w + o * 256;
            float acc = bias[o];
            for (int c = 0; c < 256; ++c) acc += wrow[c] * shiftv[c];
            const float* g = glob + b * 128;
            for (int c = 0; c < 128; ++c) acc += wrow[128 + c] * scale[128 + c] * g[c];
            rowbias[b * 128 + o] = acc;
        }
    }
}

// ============ Main GEMM: out = elu(x) @ W'^T + rowbias[graph] (+ residual) ============
// grid = 2048 blocks (128 rows each), block = 256 (8 waves)
__global__ __launch_bounds__(256)
void gemm_kernel(const float* __restrict__ x, const float* __restrict__ Wpk,
                 const float* __restrict__ rowbias, const float* __restrict__ res,
                 float* __restrict__ out)
{
    extern __shared__ float smem[];
    float* w_s  = smem;                 // 64 x 288 (padded k-pair rows)
    float* y_s  = smem + W_WORDS;       // 128 x 132 (padded activation rows)
    float* rb_s = y_s + Y_WORDS;        // 128

    const int t  = threadIdx.x;
    const int r0 = blockIdx.x * 128;    // first global row of this tile

    // stage packed weights (4096 float4)
    const float4* w4 = (const float4*)Wpk;
    #pragma unroll
    for (int i = 0; i < 16; ++i) {
        const int idx = t + 256 * i;
        const int row = idx >> 6, q = idx & 63;
        *(float4*)(w_s + row * W_STRIDE + q * 4) = w4[idx];
    }
    // stage activation tile with fused elu (4096 float4, fully coalesced)
    const float4* x4 = (const float4*)(x + (size_t)r0 * FDIM);
    #pragma unroll
    for (int i = 0; i < 16; ++i) {
        const int idx = t + 256 * i;
        const int row = idx >> 5, q = idx & 31;
        float4 v = x4[idx];
        v.x = elu(v.x); v.y = elu(v.y); v.z = elu(v.z); v.w = elu(v.w);
        *(float4*)(y_s + row * Y_STRIDE + q * 4) = v;
    }
    const int graph = blockIdx.x >> 5;  // 32 tiles per graph
    if (t < 128) rb_s[t] = rowbias[graph * 128 + t];
    __syncthreads();

    const int lane = t & 31;
    const int wv   = t >> 5;
    const int n0   = wv * 16;
    const int half = lane >> 4;
    const int l15  = lane & 15;

    v8f acc[8];
    #pragma unroll
    for (int m = 0; m < 8; ++m) acc[m] = 0.0f;

    // K loop: 32 steps of K=4; B-operand reused across the 8 M-tiles
    for (int kb = 0; kb < FDIM; kb += 4) {
        const v2f b = *(const v2f*)(w_s + ((kb >> 1) + half) * W_STRIDE + (n0 + l15) * 2);
        #pragma unroll
        for (int m = 0; m < 8; ++m) {
            const v2f a = *(const v2f*)(y_s + (m * 16 + l15) * Y_STRIDE + kb + 2 * half);
            acc[m] = __builtin_amdgcn_wmma_f32_16x16x4_f32(
                false, a, false, b, (short)0, acc[m], false, false);
        }
    }

    // Writeback per documented C layout: VGPR v, half h -> row m*16 + v + 8h, col n0+l15
    const float rb  = rb_s[n0 + l15];
    const int   col = n0 + l15;
    #pragma unroll
    for (int m = 0; m < 8; ++m) {
        const int rbase = r0 + m * 16 + half * 8;
        #pragma unroll
        for (int v = 0; v < 8; ++v) {
            const int row = rbase + v;
            float val = acc[m][v] + rb;
            if (res) val += res[row * FDIM + col];
            out[row * FDIM + col] = val;
        }
    }
}

// ============================ launcher ============================
extern "C" void kernel_launch(void* const* d_in, const int* in_sizes, int n_in,
                              void* d_out, int out_size, void* d_ws, size_t ws_size,
                              hipStream_t stream) {
    const float* inputs = (const float*)d_in[0];
    const float* mask   = (const float*)d_in[1];
    const float* gamma[2] = { (const float*)d_in[2], (const float*)d_in[6] };
    const float* beta [2] = { (const float*)d_in[3], (const float*)d_in[7] };
    const float* wmat [2] = { (const float*)d_in[4], (const float*)d_in[8] };
    const float* bvec [2] = { (const float*)d_in[5], (const float*)d_in[9] };

    float* ws = (float*)d_ws;
    float* x1      = ws;                         // 33,554,432 f (layer-0 output)
    float* S_part  = x1 + (size_t)MTOT * FDIM;   // 512*128
    float* cs_part = S_part  + RBLKS * 128;      // 512*128
    float* cq_part = cs_part + RBLKS * 128;      // 512*128
    float* mk_part = cq_part + RBLKS * 128;      // 512
    float* glob    = mk_part + RBLKS;            // 64*128
    float* scale   = glob    + BGRAPH * FDIM;    // 256
    float* shiftv  = scale   + 256;              // 256
    float* Wpk     = shiftv  + 256;              // 64*256
    float* rowbias = Wpk     + 64 * 256;         // 64*128

    float* outF = (float*)d_out;

    for (int L = 0; L < 2; ++L) {
        const float* xin = (L == 0) ? inputs : x1;
        reduce_kernel<<<RBLKS, 256, 0, stream>>>(
            xin, mask, S_part, cs_part, cq_part, mk_part, (L == 0) ? 1 : 0);
        prep_kernel<<<1, 256, 0, stream>>>(
            S_part, cs_part, cq_part, mk_part, gamma[L], beta[L], glob, scale, shiftv);
        weights_kernel<<<BGRAPH + 1, 256, 0, stream>>>(
            wmat[L], bvec[L], scale, shiftv, glob, Wpk, rowbias);
        gemm_kernel<<<MTOT / 128, 256, SMEM_BYTES, stream>>>(
            xin, Wpk, rowbias,
            (L == 1) ? inputs : nullptr,
            (L == 1) ? outF : x1);
    }
}